// CustomCNNLoss_46901042872957
// MI455X (gfx1250) — compile-verified
//
#include <hip/hip_runtime.h>
#include <stdint.h>

#define S_GRID 112
#define BATCHN 64
#define NBOX   100
#define CCLS   10
#define NCELLS (BATCHN * S_GRID * S_GRID)   // 802816
#define NBLK   3136                          // NCELLS / 256
#define NM     (BATCHN * NBOX)               // 6400

typedef float v2f  __attribute__((ext_vector_type(2)));
typedef float v8f  __attribute__((ext_vector_type(8)));
typedef unsigned int u32x4 __attribute__((ext_vector_type(4)));
typedef int   i32x4 __attribute__((ext_vector_type(4)));
typedef int   i32x8 __attribute__((ext_vector_type(8)));

__device__ __forceinline__ float sqf(float x) { return x * x; }

// ---------------------------------------------------------------------------
// Phase A: per-target decode + winner selection via packed 64-bit atomicMax.
// key = (orderedBits(best_iou) << 32) | (0xFFFFFFFF - m)
//   -> max key == max iou, tie broken toward smallest target index m,
//      exactly matching segment_max + segment_min in the reference.
// ---------------------------------------------------------------------------
__global__ void prep_targets(const float* __restrict__ pred,
                             const float* __restrict__ tgt,
                             unsigned long long* __restrict__ keys,
                             unsigned int* __restrict__ clsmask,
                             float4* __restrict__ boxvals,
                             float* __restrict__ bi,
                             int* __restrict__ bj)
{
    int m = blockIdx.x * blockDim.x + threadIdx.x;
    if (m >= NM) return;
    const float* t = tgt + (size_t)m * 5;
    float t0 = t[0];
    if (t0 < 0.0f) return;                      // invalid target
    int b = m / NBOX;
    float cx = t[1], cy = t[2], w = t[3], h = t[4];
    int col = (int)(cx * (float)S_GRID); col = min(max(col, 0), S_GRID - 1);
    int row = (int)(cy * (float)S_GRID); row = min(max(row, 0), S_GRID - 1);
    float cxr = cx * (float)S_GRID - (float)col;
    float cyr = cy * (float)S_GRID - (float)row;
    int cls = (int)t0; cls = min(max(cls, 0), CCLS - 1);
    const float cell = 1.0f / (float)S_GRID;
    const float* pc = pred + (((size_t)b * S_GRID + row) * S_GRID + col) * 20;

    float gx1 = cx - w * 0.5f, gy1 = cy - h * 0.5f;
    float gx2 = cx + w * 0.5f, gy2 = cy + h * 0.5f;
    float area_g = fmaxf(gx2 - gx1, 0.0f) * fmaxf(gy2 - gy1, 0.0f);

    float best = -1.0f; int bestj = 0;
    #pragma unroll
    for (int j = 0; j < 2; ++j) {
        float px = (pc[j * 5 + 0] + (float)col) * cell;
        float py = (pc[j * 5 + 1] + (float)row) * cell;
        float pw = pc[j * 5 + 2], ph = pc[j * 5 + 3];
        float px1 = px - pw * 0.5f, py1 = py - ph * 0.5f;
        float px2 = px + pw * 0.5f, py2 = py + ph * 0.5f;
        float iw = fmaxf(fminf(px2, gx2) - fmaxf(px1, gx1), 0.0f);
        float ih = fmaxf(fminf(py2, gy2) - fmaxf(py1, gy1), 0.0f);
        float inter = iw * ih;
        float uni = fmaxf(px2 - px1, 0.0f) * fmaxf(py2 - py1, 0.0f) + area_g - inter;
        float iou = inter / (uni + 1e-6f);
        if (iou > best) { best = iou; bestj = j; }   // strict '>' => first max, like argmax
    }

    boxvals[m] = make_float4(cxr, cyr, w, h);
    bi[m] = best;
    bj[m] = bestj;

    int cidx = (b * S_GRID + row) * S_GRID + col;
    unsigned long long key =
        ((unsigned long long)__float_as_uint(best) << 32) |
        (unsigned long long)(0xFFFFFFFFu - (unsigned)m);
    atomicMax(&keys[cidx], key);
    atomicOr(&clsmask[cidx], 1u << cls);
}

// ---------------------------------------------------------------------------
// Phase B: stream all cells. Each wave TDM-DMAs its 32-cell (2560B) slice of
// predictions into LDS (contiguous burst), waits on TENSORcnt, then computes
// the per-cell loss terms. Deterministic block reduction -> partials[block].
// ---------------------------------------------------------------------------
__global__ __launch_bounds__(256) void cell_loss(
        const float* __restrict__ pred,
        const unsigned long long* __restrict__ keys,
        const unsigned int* __restrict__ clsmask,
        const float4* __restrict__ boxvals,
        const float* __restrict__ bi,
        const int* __restrict__ bj,
        float* __restrict__ partials)
{
    __shared__ __align__(16) float smem[256 * 20];   // 20480 B staged predictions
    __shared__ float wsum[8];

    const int tid  = threadIdx.x;
    const int wave = tid >> 5;
    const int lane = tid & 31;
    const int blockCell0 = blockIdx.x * 256;

    // ---- TDM async tensor load: this wave's 2560-byte contiguous slice ----
    {
        unsigned int lds_off = (unsigned int)(uintptr_t)(&smem[0]) + (unsigned)wave * 2560u;
        unsigned long long gaddr = (unsigned long long)(uintptr_t)pred
            + ((unsigned long long)blockCell0 + (unsigned long long)wave * 32ull) * 80ull;
        const unsigned long long td0 = (unsigned long long)NCELLS * 20ull; // elements

        u32x4 g0; i32x8 g1; i32x4 g2; i32x4 g3;
        g0[0] = 1u;                                              // count=1, no gather
        g0[1] = lds_off;                                         // lds_addr (bytes)
        g0[2] = (unsigned int)(gaddr & 0xFFFFFFFFull);           // global_addr lo
        g0[3] = ((unsigned int)(gaddr >> 32) & 0x1FFFFFFu) | (2u << 30); // hi | type=2
        g1[0] = 0x00020000;                                      // data_size=2 (4B)
        g1[1] = (int)((td0 & 0xFFFFull) << 16);                  // tensor_dim0[15:0]
        g1[2] = (int)(((td0 >> 16) & 0xFFFFull) | (1u << 16));   // td0 hi | tensor_dim1=1
        g1[3] = (int)(2560u << 16);                              // tile_dim0 = 2560 elems
        g1[4] = 0;                                               // tile_dim1/2 unused
        g1[5] = (int)(td0 & 0xFFFFFFFFull);                      // dim0 stride lo
        g1[6] = 0;
        g1[7] = 0;
        g2[0] = 1; g2[1] = 0; g2[2] = 0; g2[3] = 0;
        g3[0] = 0; g3[1] = 0; g3[2] = 0; g3[3] = 0;
#if __clang_major__ >= 23
        i32x8 g4 = {0, 0, 0, 0, 0, 0, 0, 0};
        __builtin_amdgcn_tensor_load_to_lds(g0, g1, g2, g3, g4, 0);
#else
        __builtin_amdgcn_tensor_load_to_lds(g0, g1, g2, g3, 0);
#endif
        __builtin_amdgcn_s_wait_tensorcnt(0);
        asm volatile("" ::: "memory");   // LDS now holds DMA data; block reordering
    }

    // Each lane consumes its own wave's slice: cell tid, 20 floats at smem[tid*20].
    float p[20];
    {
        const float4* mp = (const float4*)(&smem[tid * 20]);
        #pragma unroll
        for (int i = 0; i < 5; ++i) {
            float4 v = mp[i];
            p[4 * i + 0] = v.x; p[4 * i + 1] = v.y;
            p[4 * i + 2] = v.z; p[4 * i + 3] = v.w;
        }
    }

    const int cell = blockCell0 + tid;
    const int b  = cell / (S_GRID * S_GRID);
    const int rc = cell - b * (S_GRID * S_GRID);
    const int r  = rc / S_GRID;
    const int c  = rc - r * S_GRID;
    const float cellw = 1.0f / (float)S_GRID;

    unsigned long long key = keys[cell];
    float loss = 0.0f;

    if (key != 0ull) {
        unsigned m = 0xFFFFFFFFu - (unsigned)(key & 0xFFFFFFFFull);
        int    j    = bj[m];
        float  biou = bi[m];
        float4 g    = boxvals[m];

        float px = p[j * 5 + 0], py = p[j * 5 + 1];
        float pw = p[j * 5 + 2], ph = p[j * 5 + 3];
        float pconf = p[j * 5 + 4];

        float pax = (px + (float)c) * cellw, pay = (py + (float)r) * cellw;
        float paw = fabsf(pw), pah = fabsf(ph);
        float gax = (g.x + (float)c) * cellw, gay = (g.y + (float)r) * cellw;
        float gaw = g.z, gah = g.w;

        const float eps = 1e-7f;
        float px1 = pax - paw * 0.5f, py1 = pay - pah * 0.5f;
        float px2 = pax + paw * 0.5f, py2 = pay + pah * 0.5f;
        float gx1 = gax - gaw * 0.5f, gy1 = gay - gah * 0.5f;
        float gx2 = gax + gaw * 0.5f, gy2 = gay + gah * 0.5f;
        float iw = fmaxf(fminf(px2, gx2) - fmaxf(px1, gx1), 0.0f);
        float ih = fmaxf(fminf(py2, gy2) - fmaxf(py1, gy1), 0.0f);
        float inter = iw * ih;
        float ap = fmaxf(px2 - px1, 0.0f) * fmaxf(py2 - py1, 0.0f);
        float ag = fmaxf(gx2 - gx1, 0.0f) * fmaxf(gy2 - gy1, 0.0f);
        float uni = ap + ag - inter;
        float iou = inter / (uni + eps);
        float rho2 = sqf(pax - gax) + sqf(pay - gay);
        float c2 = sqf(fmaxf(px2, gx2) - fminf(px1, gx1)) +
                   sqf(fmaxf(py2, gy2) - fminf(py1, gy1)) + eps;
        float v = 0.40528473456935108578f *     // 4/pi^2
                  sqf(atanf(gaw / (gah + eps)) - atanf(paw / (pah + eps)));
        float alpha = v / (1.0f - iou + v + eps);
        float ciou = 1.0f - iou + rho2 / c2 + alpha * v;

        loss += 5.0f * ciou;                        // LAMBDA_COORD
        loss += sqf(pconf - biou);                  // obj
        float oconf = p[(1 - j) * 5 + 4];
        loss += 0.1f * (oconf * oconf);             // LAMBDA_NOOBJ, non-resp box

        unsigned cm = clsmask[cell];
        #pragma unroll
        for (int k = 0; k < CCLS; ++k) {            // focal loss on 10 classes
            float l  = p[10 + k];
            float tt = (float)((cm >> k) & 1u);
            float bce = fmaxf(l, 0.0f) - l * tt + log1pf(expf(-fabsf(l)));
            float pr  = 1.0f / (1.0f + expf(-l));
            float pt  = tt * pr + (1.0f - tt) * (1.0f - pr);
            float at  = tt * 0.25f + (1.0f - tt) * 0.75f;
            loss += at * sqf(1.0f - pt) * bce;
        }
    } else {
        loss += 0.1f * (p[4] * p[4] + p[9] * p[9]); // noobj, both boxes
    }

    // Deterministic block reduction (wave32 shuffle tree + LDS across 8 waves).
    #pragma unroll
    for (int off = 16; off >= 1; off >>= 1) loss += __shfl_xor(loss, off, 32);
    if (lane == 0) wsum[wave] = loss;
    __syncthreads();
    if (tid == 0) {
        float s = 0.0f;
        #pragma unroll
        for (int i = 0; i < 8; ++i) s += wsum[i];
        partials[blockIdx.x] = s;
    }
}

// ---------------------------------------------------------------------------
// Phase C: one wave reduces 3136 block partials. 64 lane accumulators are
// collapsed on the matrix pipe: D = ones(16x4) x B(4x16) + 0 gives column
// sums of B in every row of D (f32 end to end), then a 5-step shuffle.
// ---------------------------------------------------------------------------
__global__ void final_reduce(const float* __restrict__ partials,
                             float* __restrict__ out)
{
    int lane = threadIdx.x;          // 32 lanes
    v2f acc = {0.0f, 0.0f};
    for (int base = 2 * lane; base < NBLK; base += 64) {
        acc.x += partials[base];
        acc.y += partials[base + 1];
    }
    v2f ones = {1.0f, 1.0f};
    v8f cz = {};
    v8f d = __builtin_amdgcn_wmma_f32_16x16x4_f32(
        /*neg_a=*/false, ones, /*neg_b=*/false, acc,
        /*c_mod=*/(short)0, cz, /*reuse_a=*/false, /*reuse_b=*/false);
    float r = d[0];                  // lanes hold column sums (each appears twice)
    #pragma unroll
    for (int off = 16; off >= 1; off >>= 1) r += __shfl_xor(r, off, 32);
    if (lane == 0) out[0] = r * 0.5f * (1.0f / (float)BATCHN);
}

// ---------------------------------------------------------------------------
extern "C" void kernel_launch(void* const* d_in, const int* in_sizes, int n_in,
                              void* d_out, int out_size, void* d_ws, size_t ws_size,
                              hipStream_t stream)
{
    (void)in_sizes; (void)n_in; (void)out_size; (void)ws_size;
    const float* pred = (const float*)d_in[0];
    const float* tgt  = (const float*)d_in[1];

    char* ws = (char*)d_ws;
    const size_t NC = (size_t)NCELLS;
    unsigned long long* keys    = (unsigned long long*)ws;            // 8*NC
    unsigned int*       clsmask = (unsigned int*)(ws + 8 * NC);       // 4*NC
    float4* boxvals = (float4*)(ws + 12 * NC);                        // 16*NM
    float*  bi      = (float*) (ws + 12 * NC + 16 * (size_t)NM);      // 4*NM
    int*    bj      = (int*)   (ws + 12 * NC + 20 * (size_t)NM);      // 4*NM
    float*  partials= (float*) (ws + 12 * NC + 24 * (size_t)NM);      // 4*NBLK

    hipMemsetAsync(ws, 0, 12 * NC, stream);                  // keys + clsmask
    prep_targets<<<(NM + 255) / 256, 256, 0, stream>>>(pred, tgt, keys, clsmask,
                                                       boxvals, bi, bj);
    cell_loss<<<NBLK, 256, 0, stream>>>(pred, keys, clsmask, boxvals, bi, bj,
                                        partials);
    final_reduce<<<1, 32, 0, stream>>>(partials, (float*)d_out);
}